// SmoothAP_Cross_81475529605816
// MI455X (gfx1250) — compile-verified
//
#include <hip/hip_runtime.h>
#include <math.h>

#define BATCH 384
#define FEAT  256
#define GROUP 8
#define NUM_ID 48

typedef __attribute__((ext_vector_type(2))) float v2f;
typedef __attribute__((ext_vector_type(8))) float v8f;

// _temp_sigmoid(t, 0.01) == 1 / (1 + exp(clip(-100*t, -50, 50)))
__device__ __forceinline__ float sigmoid_anneal(float t) {
    float e = fminf(fmaxf(-100.0f * t, -50.0f), 50.0f);
    return 1.0f / (1.0f + __expf(e));
}

// One wave (32 lanes) per row; rows [0,384) -> preds, [384,768) -> gallery.
__global__ void rownorm_kernel(const float* __restrict__ preds,
                               const float* __restrict__ gallery,
                               float* __restrict__ inv_p,
                               float* __restrict__ inv_g) {
    const int r    = blockIdx.x;
    const int lane = threadIdx.x;
    const float* src = (r < BATCH) ? (preds + r * FEAT)
                                   : (gallery + (r - BATCH) * FEAT);
    float s = 0.0f;
    #pragma unroll
    for (int t = 0; t < FEAT / 32; ++t) {
        float x = src[lane + t * 32];
        s += x * x;
    }
    #pragma unroll
    for (int m = 16; m >= 1; m >>= 1) s += __shfl_xor(s, m, 32);
    if (lane == 0) {
        float inv = 1.0f / fmaxf(sqrtf(s), 1e-12f);
        if (r < BATCH) inv_p[r] = inv; else inv_g[r - BATCH] = inv;
    }
}

// One wave per 16x16 tile of sim = (preds @ gallery^T), scaled by inv norms
// at store time. Uses V_WMMA_F32_16X16X4_F32, K-loop over FEAT in steps of 4.
// A (16x4 f32) fragment layout: lane l holds M = l&15, K = (l>>4)*2 + {0,1}.
// B (4x16) fragment is symmetric with N = l&15.
__global__ void simgemm_kernel(const float* __restrict__ preds,
                               const float* __restrict__ gallery,
                               const float* __restrict__ inv_p,
                               const float* __restrict__ inv_g,
                               float* __restrict__ sim) {
    const int bm   = blockIdx.x;
    const int bn   = blockIdx.y;
    const int lane = threadIdx.x;
    const int mn   = lane & 15;
    const int koff = (lane >> 4) << 1;   // 0 or 2

    const float* __restrict__ pA = preds   + (bm * 16 + mn) * FEAT + koff;
    const float* __restrict__ pB = gallery + (bn * 16 + mn) * FEAT + koff;

    v8f c = {};
    #pragma unroll 8
    for (int kk = 0; kk < FEAT; kk += 4) {
        v2f a = *(const v2f*)(pA + kk);   // 8B aligned (koff even)
        v2f b = *(const v2f*)(pB + kk);
        c = __builtin_amdgcn_wmma_f32_16x16x4_f32(
                /*neg_a=*/false, a, /*neg_b=*/false, b,
                /*c_mod=*/(short)0, c, /*reuse_a=*/false, /*reuse_b=*/false);
    }

    // C/D layout: VGPR r -> M = r + (lane>>4)*8, N = lane&15.
    const int col   = bn * 16 + mn;
    const float sg  = inv_g[col];
    const int rbase = bm * 16 + ((lane >> 4) << 3);
    #pragma unroll
    for (int r = 0; r < 8; ++r) {
        const int row = rbase + r;
        sim[row * BATCH + col] = c[r] * inv_p[row] * sg;
    }
}

// One wave per row i. Only the 8 in-group rank sums rk[i, j] are ever used
// downstream (diagonal blocks), so compute exactly those: each lane streams
// the row once and accumulates 8 sigmoid sums simultaneously.
__global__ void rank_kernel(const float* __restrict__ sim,
                            float* __restrict__ contrib) {
    const int i     = blockIdx.x;
    const int lane  = threadIdx.x;
    const int gbase = (i / GROUP) * GROUP;
    const float* __restrict__ row = sim + i * BATCH;

    float d[GROUP];
    #pragma unroll
    for (int j = 0; j < GROUP; ++j) d[j] = row[gbase + j];

    float acc[GROUP];
    #pragma unroll
    for (int j = 0; j < GROUP; ++j) acc[j] = 0.0f;

    #pragma unroll 2
    for (int k = lane; k < BATCH; k += 32) {
        const float s = row[k];
        #pragma unroll
        for (int j = 0; j < GROUP; ++j) acc[j] += sigmoid_anneal(s - d[j]);
    }

    #pragma unroll
    for (int j = 0; j < GROUP; ++j) {
        #pragma unroll
        for (int m = 16; m >= 1; m >>= 1) acc[j] += __shfl_xor(acc[j], m, 32);
    }

    if (lane == 0) {
        float total = 0.0f;
        #pragma unroll
        for (int j = 0; j < GROUP; ++j) {
            float pos = 0.0f;
            #pragma unroll
            for (int c = 0; c < GROUP; ++c) pos += sigmoid_anneal(d[c] - d[j]);
            total += pos / acc[j];
        }
        contrib[i] = total;
    }
}

__global__ void finalize_kernel(const float* __restrict__ contrib,
                                float* __restrict__ out) {
    const int lane = threadIdx.x;
    float s = 0.0f;
    #pragma unroll
    for (int k = lane; k < BATCH; k += 32) s += contrib[k];
    #pragma unroll
    for (int m = 16; m >= 1; m >>= 1) s += __shfl_xor(s, m, 32);
    if (lane == 0) out[0] = 1.0f - s * (1.0f / (float)(GROUP * BATCH));
}

extern "C" void kernel_launch(void* const* d_in, const int* in_sizes, int n_in,
                              void* d_out, int out_size, void* d_ws, size_t ws_size,
                              hipStream_t stream) {
    (void)in_sizes; (void)n_in; (void)out_size; (void)ws_size;
    const float* preds   = (const float*)d_in[0];
    const float* gallery = (const float*)d_in[1];
    float* ws      = (float*)d_ws;
    float* inv_p   = ws;                         // 384
    float* inv_g   = ws + BATCH;                 // 384
    float* sim     = ws + 2 * BATCH;             // 384*384
    float* contrib = sim + BATCH * BATCH;        // 384
    float* out     = (float*)d_out;

    rownorm_kernel<<<2 * BATCH, 32, 0, stream>>>(preds, gallery, inv_p, inv_g);
    simgemm_kernel<<<dim3(BATCH / 16, BATCH / 16), 32, 0, stream>>>(
        preds, gallery, inv_p, inv_g, sim);
    rank_kernel<<<BATCH, 32, 0, stream>>>(sim, contrib);
    finalize_kernel<<<1, 32, 0, stream>>>(contrib, out);
}